// DFTD3_17085379903617
// MI455X (gfx1250) — compile-verified
//
#include <hip/hip_runtime.h>
#include <stdint.h>
#include <stddef.h>

// ---------------- problem constants (fixed by the reference) ----------------
constexpr int   B = 32, N = 256, M = 128;
constexpr float S6f = 1.0f, S8f = 1.2177f, A1f = 0.4145f, A2f = 4.8593f;
constexpr float BOHR_INVf = 1.8897261254578281f;
constexpr float HALF_HARTREEf = 13.605693012183622f;
constexpr float LOG2E = 1.4426950408889634f;
constexpr float K1L2 = -16.0f * LOG2E;   // exp(K1*x) == exp2(K1L2*x)
constexpr float K3L2 = -4.0f  * LOG2E;   // exp(K3*x) == exp2(K3L2*x)

// ---------------- CDNA5 async global->LDS copy helpers ----------------------
// Each active lane copies 16B from the global address in its vaddr to the LDS address in its vdst. Tracked by
// ASYNCcnt; wait with s_wait_asynccnt then barrier before any wave reads LDS.
__device__ __forceinline__ void async_ld_b128(void* lds_dst, const void* gsrc) {
    asm volatile("global_load_async_to_lds_b128 %0, %1, off"
                 :
                 : "v"((uint32_t)(uintptr_t)lds_dst), "v"(gsrc)
                 : "memory");
}
__device__ __forceinline__ void async_wait0() {
    asm volatile("s_wait_asynccnt 0" ::: "memory");
}

// wave32 deterministic tree sum
__device__ __forceinline__ float wave_sum(float v) {
#pragma unroll
    for (int off = 16; off; off >>= 1) v += __shfl_down(v, off, 32);
    return v;
}

// ============================================================================
// Kernel 1: coordination numbers.  One 128-thread block per (b, n); thread t
// handles neighbor m = t.  Writes ws_cn[b,n] and ws_rr[b,n] = r4r2[z_i].
// ============================================================================
__global__ __launch_bounds__(128) void d3_cn_kernel(
    const int* __restrict__ numbers, const int* __restrict__ nbr_idx,
    const unsigned char* __restrict__ pair_mask, const float* __restrict__ d_ij,
    const float* __restrict__ r4r2, const float* __restrict__ rcov,
    const float* __restrict__ cnmax,
    float* __restrict__ ws_cn, float* __restrict__ ws_rr) {
    __shared__ __align__(16) int s_num[N];
    __shared__ float s_red[4];
    const int n = blockIdx.x, b = blockIdx.y, t = threadIdx.x;
    const size_t row = (size_t)b * N;

    if (t < 64) async_ld_b128(&s_num[t * 4], numbers + row + t * 4);
    async_wait0();
    __syncthreads();

    const size_t base = (row + n) * (size_t)M;
    float cn_ij = 0.0f;
    if (pair_mask[base + t]) {
        const float d   = d_ij[base + t] * BOHR_INVf;
        const int   nbr = nbr_idx[base + t];
        const float rc  = rcov[s_num[n]] + rcov[s_num[nbr]];
        const float e   = __builtin_amdgcn_exp2f(
            K1L2 * (rc * __builtin_amdgcn_rcpf(d) - 1.0f));
        cn_ij = __builtin_amdgcn_rcpf(1.0f + e);
    }
    const float v = wave_sum(cn_ij);
    if ((t & 31) == 0) s_red[t >> 5] = v;
    __syncthreads();
    if (t == 0) {
        const int zi = s_num[n];
        float cn = s_red[0] + s_red[1] + s_red[2] + s_red[3];
        cn = fminf(cn, cnmax[zi]);
        ws_cn[row + n] = cn;
        ws_rr[row + n] = r4r2[zi];
    }
}

// ============================================================================
// Kernel 2: per-pair C6 interpolation (5x5 references, 25 exp2) + damped
// energy.  Per-batch rows (numbers/cn/rr) async-staged into LDS so the three
// per-pair gathers are LDS lookups.  Writes one partial sum per (b, n).
// ============================================================================
__global__ __launch_bounds__(128) void d3_energy_kernel(
    const int* __restrict__ numbers, const int* __restrict__ nbr_idx,
    const unsigned char* __restrict__ pair_mask, const float* __restrict__ d_ij,
    const float* __restrict__ c6ab,
    const float* __restrict__ ws_cn, const float* __restrict__ ws_rr,
    float* __restrict__ partial) {
    __shared__ __align__(16) int   s_num[N];
    __shared__ __align__(16) float s_cn[N];
    __shared__ __align__(16) float s_rr[N];
    __shared__ float s_red[4];
    const int n = blockIdx.x, b = blockIdx.y, t = threadIdx.x;
    const size_t row = (size_t)b * N;

    if (t < 64) {
        async_ld_b128(&s_num[t * 4], numbers + row + t * 4);
        async_ld_b128(&s_rr[t * 4],  ws_rr   + row + t * 4);
    } else {
        const int u = t - 64;
        async_ld_b128(&s_cn[u * 4], ws_cn + row + u * 4);
    }
    async_wait0();
    __syncthreads();

    const size_t base = (row + n) * (size_t)M;
    float e = 0.0f;
    if (pair_mask[base + t]) {               // masked pairs contribute exactly 0
        const int   nbr = nbr_idx[base + t];
        const float d   = d_ij[base + t] * BOHR_INVf;
        const int   zi  = s_num[n], zj = s_num[nbr];
        const float cni = s_cn[n],  cnj = s_cn[nbr];
        const float rri = s_rr[n],  rrj = s_rr[nbr];

        const float* __restrict__ cp = c6ab + (size_t)(zi * 95 + zj) * 75;
        float w = 0.0f, z = 0.0f;
#pragma unroll
        for (int r = 0; r < 25; ++r) {
            const float c6r = cp[3 * r + 0];
            const float ci  = cp[3 * r + 1];
            const float cj  = cp[3 * r + 2];
            const float u0  = cni - ci;
            const float u1  = cnj - cj;
            const float l   = __builtin_amdgcn_exp2f(
                K3L2 * fmaf(u0, u0, u1 * u1));
            w += l;
            z = fmaf(c6r, l, z);
        }
        const float c6 = (w < 1e-5f) ? 0.0f : z * __builtin_amdgcn_rcpf(w);

        const float rrij = 3.0f * rri * rrj;
        const float r0   = fmaf(A1f, __builtin_amdgcn_sqrtf(rrij), A2f);
        const float d2 = d * d,   d4 = d2 * d2, d6 = d4 * d2, d8 = d4 * d4;
        const float p2 = r0 * r0, p4 = p2 * p2, p6 = p4 * p2, p8 = p4 * p4;
        e = c6 * (S6f * __builtin_amdgcn_rcpf(d6 + p6)
                + S8f * rrij * __builtin_amdgcn_rcpf(d8 + p8));
    }
    const float v = wave_sum(e);
    if ((t & 31) == 0) s_red[t >> 5] = v;
    __syncthreads();
    if (t == 0) partial[row + n] = s_red[0] + s_red[1] + s_red[2] + s_red[3];
}

// ============================================================================
// Kernel 3: deterministic per-batch reduction of the N partials.
// ============================================================================
__global__ __launch_bounds__(256) void d3_reduce_kernel(
    const float* __restrict__ partial, float* __restrict__ out) {
    __shared__ float s_red[8];
    const int b = blockIdx.x, t = threadIdx.x;
    const float v = wave_sum(partial[(size_t)b * N + t]);
    if ((t & 31) == 0) s_red[t >> 5] = v;
    __syncthreads();
    if (t == 0) {
        float s = 0.0f;
#pragma unroll
        for (int i = 0; i < 8; ++i) s += s_red[i];
        out[b] = -HALF_HARTREEf * s;
    }
}

// ============================================================================
extern "C" void kernel_launch(void* const* d_in, const int* in_sizes, int n_in,
                              void* d_out, int out_size, void* d_ws, size_t ws_size,
                              hipStream_t stream) {
    (void)in_sizes; (void)n_in; (void)out_size; (void)ws_size;
    // setup_inputs order:
    const int*           numbers   = (const int*)d_in[0];            // (B,N) int32
    const int*           nbr_idx   = (const int*)d_in[1];            // (B,N,M) int32
    const unsigned char* pair_mask = (const unsigned char*)d_in[2];  // (B,N,M) bool
    const float*         d_ij      = (const float*)d_in[3];          // (B,N,M) f32
    const float*         c6ab      = (const float*)d_in[4];          // (95,95,5,5,3)
    const float*         r4r2      = (const float*)d_in[5];          // (95,)
    const float*         rcov      = (const float*)d_in[6];          // (95,)
    const float*         cnmax     = (const float*)d_in[7];          // (95,)
    float* out = (float*)d_out;

    // workspace: cn[B*N] | rr[B*N] | partial[B*N]  (96 KB total)
    float* ws_cn   = (float*)d_ws;
    float* ws_rr   = ws_cn + (size_t)B * N;
    float* partial = ws_rr + (size_t)B * N;

    dim3 grid(N, B);
    d3_cn_kernel<<<grid, 128, 0, stream>>>(numbers, nbr_idx, pair_mask, d_ij,
                                           r4r2, rcov, cnmax, ws_cn, ws_rr);
    d3_energy_kernel<<<grid, 128, 0, stream>>>(numbers, nbr_idx, pair_mask, d_ij,
                                               c6ab, ws_cn, ws_rr, partial);
    d3_reduce_kernel<<<B, 256, 0, stream>>>(partial, out);
}